// CoordinateDescentRouter_55044300865624
// MI455X (gfx1250) — compile-verified
//
#include <hip/hip_runtime.h>
#include <hip/hip_bf16.h>
#include <stdint.h>

typedef float v2f __attribute__((ext_vector_type(2)));
typedef float v8f __attribute__((ext_vector_type(8)));

#define D_DIM   2048
#define N_PER_B 8192
#define N_BATCH 4
#define N_ROWS  (N_PER_B * N_BATCH)
#define NUM_TOK 1024
#define EFF_K   1152.0f     /* min(1024 * 9/8, 8192) */
#define N_ITERS 50

// ---------------------------------------------------------------------------
// Kernel A: s[row] = dot(x[row,:], r[:]) via V_WMMA_F32_16X16X4_F32.
// Each wave owns a 16-row tile; B fragment = routing token replicated across
// all 16 N columns, so every D column carries the same 16 row-sums.
// HBM-bound: streams x (256 MiB) exactly once.
// ---------------------------------------------------------------------------
__global__ __launch_bounds__(256)
void router_matvec_wmma(const float* __restrict__ x,
                        const float* __restrict__ rtok,
                        float* __restrict__ s_out)
{
    __shared__ float rl[D_DIM];
    const int t = threadIdx.x;
#pragma unroll
    for (int i = 0; i < D_DIM / 256; ++i) rl[t + i * 256] = rtok[t + i * 256];
    __syncthreads();

    const int lane = t & 31;
    const int wave = t >> 5;
    const int half = lane >> 4;          // 0: K pair (0,1); 1: K pair (2,3)
    const int mrow = lane & 15;
    const int tileBase = blockIdx.x * 128 + wave * 16;

    const float* xrow = x + (size_t)(tileBase + mrow) * D_DIM + 2 * half;
    const float* rptr = rl + 2 * half;

    v8f acc = {};
    for (int dB = 0; dB < D_DIM; dB += 128) {
#pragma unroll
        for (int d0 = 0; d0 < 128; d0 += 4) {
            v2f av = *(const v2f*)(xrow + dB + d0);   // A: 16x4 f32 fragment
            v2f bv = *(const v2f*)(rptr + dB + d0);   // B: 4x16, all N cols = r
            acc = __builtin_amdgcn_wmma_f32_16x16x4_f32(
                false, av, false, bv, (short)0, acc, false, false);
        }
    }

    // All N columns identical: lane 0 holds rows 0..7 (VGPR v = row v),
    // lane 16 holds rows 8..15.
    if (lane == 0) {
#pragma unroll
        for (int m = 0; m < 8; ++m) s_out[tileBase + m] = acc[m];
    } else if (lane == 16) {
#pragma unroll
        for (int m = 0; m < 8; ++m) s_out[tileBase + 8 + m] = acc[m];
    }
}

// ---------------------------------------------------------------------------
// Kernel B: per score-row (4 total): coordinate descent (50 iters, algebraic
// min(e,E) form) + bitonic top-k ordering in LDS.
// ---------------------------------------------------------------------------
__global__ __launch_bounds__(1024)
void router_descent_topk(const float* __restrict__ s_in,
                         float* __restrict__ sel_scores,
                         int* __restrict__ sel_idx)
{
    __shared__ unsigned int   kkey[N_PER_B];    // 32 KB
    __shared__ unsigned short kidx[N_PER_B];    // 16 KB
    __shared__ float red[32];
    __shared__ float bcast;

    const int row  = blockIdx.x;
    const int t    = threadIdx.x;
    const int lane = t & 31;
    const int wave = t >> 5;
    const float* srow = s_in + (size_t)row * N_PER_B;

    float sv[8];
#pragma unroll
    for (int k = 0; k < 8; ++k) sv[k] = srow[t + k * 1024];

    // ---- row max M (for stable exp) ----
    float m = sv[0];
#pragma unroll
    for (int k = 1; k < 8; ++k) m = fmaxf(m, sv[k]);
#pragma unroll
    for (int off = 16; off > 0; off >>= 1) m = fmaxf(m, __shfl_xor(m, off, 32));
    if (lane == 0) red[wave] = m;
    __syncthreads();
    if (t == 0) {
        float mm = red[0];
        for (int i = 1; i < 32; ++i) mm = fmaxf(mm, red[i]);
        bcast = mm;
    }
    __syncthreads();
    const float M = bcast;

    float ev[8];
#pragma unroll
    for (int k = 0; k < 8; ++k) ev[k] = __expf(sv[k] - M);

    const float logk = __logf(EFF_K);
    float a = logk - __logf((float)N_PER_B);   // closed-form iteration 1

    // Iterations 2..50: a <- logk - M - log( sum_j min(e_j, exp(-a-M)) )
    for (int it = 1; it < N_ITERS; ++it) {
        const float E = __expf(-a - M);
        float p = 0.0f;
#pragma unroll
        for (int k = 0; k < 8; ++k) p += fminf(ev[k], E);
#pragma unroll
        for (int off = 16; off > 0; off >>= 1) p += __shfl_xor(p, off, 32);
        if (lane == 0) red[wave] = p;
        __syncthreads();
        if (t == 0) {
            float sum = red[0];
            for (int i = 1; i < 32; ++i) sum += red[i];
            bcast = logk - M - __logf(sum);
        }
        __syncthreads();
        a = bcast;
    }

    // ---- build sort keys: score order == order of min(s+a, 0); ties (exact
    // 0 => score 1.0) broken by ascending index to mimic jax's stable top_k.
#pragma unroll
    for (int k = 0; k < 8; ++k) {
        const int j = t + k * 1024;
        const float keyf = fminf(sv[k] + a, 0.0f);
        unsigned u = __float_as_uint(keyf);
        u = (u & 0x80000000u) ? ~u : (u | 0x80000000u);   // monotone f32->u32
        kkey[j] = u;
        kidx[j] = (unsigned short)j;
    }
    __syncthreads();

    // ---- bitonic sort, descending key / ascending index on ties ----
    for (unsigned size = 2; size <= (unsigned)N_PER_B; size <<= 1) {
        for (unsigned stride = size >> 1; stride > 0; stride >>= 1) {
#pragma unroll
            for (unsigned v = t; v < N_PER_B / 2; v += 1024) {
                const unsigned i = 2u * v - (v & (stride - 1u));
                const unsigned j = i + stride;
                unsigned ki = kkey[i], kj = kkey[j];
                unsigned short ii = kidx[i], ij = kidx[j];
                // jFirst: element j should precede i in (desc key, asc idx)
                const bool jFirst = (kj > ki) || (kj == ki && ij < ii);
                const bool descBlock = ((i & size) == 0u);
                if (descBlock ? jFirst : !jFirst) {
                    kkey[i] = kj; kkey[j] = ki;
                    kidx[i] = ij; kidx[j] = ii;
                }
            }
            __syncthreads();
        }
    }

    // ---- emit: straight-through scores are exactly 1.0 ----
    if (t < NUM_TOK) {
        sel_scores[row * NUM_TOK + t] = 1.0f;
        sel_idx[row * NUM_TOK + t]    = (int)kidx[t];
    }
}

// ---------------------------------------------------------------------------
extern "C" void kernel_launch(void* const* d_in, const int* in_sizes, int n_in,
                              void* d_out, int out_size, void* d_ws, size_t ws_size,
                              hipStream_t stream)
{
    (void)in_sizes; (void)n_in; (void)out_size; (void)ws_size;

    const float* x    = (const float*)d_in[0];
    const float* rtok = (const float*)d_in[1];

    float* s          = (float*)d_ws;                       // 32768 f32 scratch
    float* sel_scores = (float*)d_out;                      // [4,1024] f32
    int*   sel_idx    = (int*)((float*)d_out + N_BATCH * NUM_TOK); // [4,1024] i32

    router_matvec_wmma<<<dim3(N_ROWS / 128), dim3(256), 0, stream>>>(x, rtok, s);
    router_descent_topk<<<dim3(N_BATCH), dim3(1024), 0, stream>>>(s, sel_scores, sel_idx);
}